// MyGraphSage_11622181503640
// MI455X (gfx1250) — compile-verified
//
#include <hip/hip_runtime.h>

typedef __attribute__((ext_vector_type(2))) float v2f;
typedef __attribute__((ext_vector_type(8))) float v8f;

#define N_NODES 100000
#define N_EDGES 1600000
#define D       64
#define N_TILES (N_NODES / 16)   // 6250, exact
#define NEG_SLOPE 0.01f

// LDS layout for W: 32 K-pairs, each row holds 64 interleaved {W[2p][n], W[2p+1][n]}
// pairs (128 floats) padded to stride 160 (160 % 64 == 32 -> lane-halves hit
// disjoint LDS bank halves).
#define WP_STRIDE 160
#define WP_ELEMS  (32 * WP_STRIDE)   // 5120 floats = 20 KB

// Guaranteed hardware fp32 atomic add (no CAS fallback).
__device__ __forceinline__ void atomic_add_f32(float* p, float v) {
    asm volatile("global_atomic_add_f32 %0, %1, off" :: "v"(p), "v"(v) : "memory");
}

// acc[v][:] = feat[v][:]  (folds the "+ feat_v" self term), deg[v] = 1.0 (folds "+1")
__global__ void sage_init_kernel(const float* __restrict__ feat,
                                 float* __restrict__ acc,
                                 float* __restrict__ deg) {
    int tid = blockIdx.x * blockDim.x + threadIdx.x;
    if (tid < N_NODES * D) acc[tid] = feat[tid];
    if (tid < N_NODES)     deg[tid] = 1.0f;
}

// One thread per (edge, 4 columns): gather feat[src] (float4), scatter-add into acc[dst].
__global__ void sage_scatter_kernel(const float* __restrict__ feat,
                                    const int* __restrict__ src,
                                    const int* __restrict__ dst,
                                    float* __restrict__ acc,
                                    float* __restrict__ deg) {
    int tid = blockIdx.x * blockDim.x + threadIdx.x;
    if (tid >= N_EDGES * 16) return;
    int e  = tid >> 4;
    int c4 = (tid & 15) << 2;
    int s = src[e];
    int d = dst[e];
    const float4 f = *(const float4*)(feat + s * D + c4);
    float* base = acc + (size_t)d * D + c4;
    atomic_add_f32(base + 0, f.x);
    atomic_add_f32(base + 1, f.y);
    atomic_add_f32(base + 2, f.z);
    atomic_add_f32(base + 3, f.w);
    if ((tid & 15) == 0) atomic_add_f32(deg + d, 1.0f);
}

// One wave (32 lanes) per 16-node tile. D = (acc/deg) @ W using V_WMMA_F32_16X16X4_F32.
__global__ void __launch_bounds__(256)
sage_wmma_kernel(const float* __restrict__ acc,
                 const float* __restrict__ deg,
                 const float* __restrict__ W,     // [64 in][64 out], row-major
                 const float* __restrict__ bias,  // [64]
                 float* __restrict__ out) {
    __shared__ float Wlds[WP_ELEMS];  // pair-interleaved W, 20 KB

    // Stage W pair-interleaved: Wlds[p*WP_STRIDE + 2n + {0,1}] = W[2p][n], W[2p+1][n].
    // Consecutive threads -> consecutive n: both global reads coalesce per row.
    for (int i = threadIdx.x; i < 32 * 64; i += 256) {
        int p = i >> 6;          // K-pair index 0..31
        int n = i & 63;          // output column
        float2 w2;
        w2.x = W[(2 * p)     * D + n];
        w2.y = W[(2 * p + 1) * D + n];
        *(float2*)&Wlds[p * WP_STRIDE + 2 * n] = w2;
    }
    __syncthreads();

    const int wave = threadIdx.x >> 5;
    const int lane = threadIdx.x & 31;
    const int tile = blockIdx.x * 8 + wave;
    if (tile >= N_TILES) return;   // wave-uniform: EXEC stays all-1s for WMMA

    const int row = lane & 15;     // M (for A/C halves) and N (for B/C column)
    const int hi  = lane >> 4;     // 0: lanes 0-15, 1: lanes 16-31
    const int node = tile * 16 + row;

    const float  dscale = 1.0f / deg[node];          // deg already includes +1
    const float* hrow   = acc + (size_t)node * D;

    v8f c0 = {}, c1 = {}, c2 = {}, c3 = {};

    #pragma unroll
    for (int kb = 0; kb < 16; ++kb) {
        const int k0 = kb * 4 + hi * 2;
        // A 16x4 f32 frag: VGPR0 = A[M=row][k0], VGPR1 = A[M=row][k0+1]
        v2f a = *(const v2f*)(hrow + k0);
        a.x *= dscale;
        a.y *= dscale;
        // B 4x16 f32 frags, one contiguous b64 LDS load each:
        // pair index = kb*2 + hi, element = {W[k0][N], W[k0+1][N]}, N = nb*16 + row
        const float* wb = &Wlds[(kb * 2 + hi) * WP_STRIDE + 2 * row];
        v2f b0 = *(const v2f*)(wb +  0);
        v2f b1 = *(const v2f*)(wb + 32);
        v2f b2 = *(const v2f*)(wb + 64);
        v2f b3 = *(const v2f*)(wb + 96);
        c0 = __builtin_amdgcn_wmma_f32_16x16x4_f32(false, a, false, b0, (short)0, c0, false, false);
        c1 = __builtin_amdgcn_wmma_f32_16x16x4_f32(false, a, false, b1, (short)0, c1, false, false);
        c2 = __builtin_amdgcn_wmma_f32_16x16x4_f32(false, a, false, b2, (short)0, c2, false, false);
        c3 = __builtin_amdgcn_wmma_f32_16x16x4_f32(false, a, false, b3, (short)0, c3, false, false);
    }

    // Epilogue: bias + leaky-relu + store.
    // C layout: VGPR j, lanes 0-15 -> M=j, lanes 16-31 -> M=j+8; col = lane&15.
    const float bb0 = bias[row], bb1 = bias[16 + row], bb2 = bias[32 + row], bb3 = bias[48 + row];
    const int mbase = tile * 16 + hi * 8;
    #pragma unroll
    for (int j = 0; j < 8; ++j) {
        float* orow = out + (size_t)(mbase + j) * D;
        float v;
        v = c0[j] + bb0; orow[row +  0] = (v >= 0.f) ? v : NEG_SLOPE * v;
        v = c1[j] + bb1; orow[row + 16] = (v >= 0.f) ? v : NEG_SLOPE * v;
        v = c2[j] + bb2; orow[row + 32] = (v >= 0.f) ? v : NEG_SLOPE * v;
        v = c3[j] + bb3; orow[row + 48] = (v >= 0.f) ? v : NEG_SLOPE * v;
    }
}

extern "C" void kernel_launch(void* const* d_in, const int* in_sizes, int n_in,
                              void* d_out, int out_size, void* d_ws, size_t ws_size,
                              hipStream_t stream) {
    const float* feat = (const float*)d_in[0];
    const int*   src  = (const int*)d_in[1];
    const int*   dst  = (const int*)d_in[2];
    const float* W    = (const float*)d_in[3];
    const float* bias = (const float*)d_in[4];
    float*       out  = (float*)d_out;

    float* acc = (float*)d_ws;                 // N_NODES * 64 floats = 25.6 MB
    float* deg = acc + (size_t)N_NODES * D;    // N_NODES floats

    {   // init
        int n = N_NODES * D;
        sage_init_kernel<<<(n + 255) / 256, 256, 0, stream>>>(feat, acc, deg);
    }
    {   // scatter-add over edges
        int n = N_EDGES * 16;
        sage_scatter_kernel<<<(n + 255) / 256, 256, 0, stream>>>(feat, src, dst, acc, deg);
    }
    {   // normalize + GEMM + bias + leaky-relu via f32 WMMA
        int blocks = (N_TILES + 7) / 8;        // 8 waves (tiles) per 256-thread block
        sage_wmma_kernel<<<blocks, 256, 0, stream>>>(acc, deg, W, bias, out);
    }
}